// AdaptiveQuantizer_463856468234
// MI455X (gfx1250) — compile-verified
//
#include <hip/hip_runtime.h>

// AdaptiveQuantizer for MI455X (gfx1250).
//
// Shapes fixed by the reference:
//   features: (B=16, C=64, H=256, W=256) float32
//   bits:     (B, H, W) int32 in [1,8]
//   out:      same as features
//
// Roofline: 268.4 MB in + 268.4 MB out + 4 MB bits = ~541 MB total
// -> ~23 us at 23.3 TB/s HBM. Compute (~0.5 G VALU ops) is negligible.
// Strategy: single pass, one thread per pixel, all 64 channel values held
// in registers so features are read from HBM exactly once. Channel stride
// is H*W elements, but 32 consecutive lanes cover 32 consecutive w, so
// every load/store in the C-loop is a fully coalesced 128B cacheline.
// Non-temporal hints on the streaming loads/stores keep L2 clean.

constexpr int   kB    = 16;
constexpr int   kC    = 64;
constexpr int   kH    = 256;
constexpr int   kW    = 256;
constexpr int   kHW   = kH * kW;        // 65536
constexpr int   kNPix = kB * kHW;       // 1,048,576 pixels
constexpr float kEps  = 1e-8f;

__global__ __launch_bounds__(256) void AdaptiveQuantizer_463856468234_kernel(
    const float* __restrict__ features,
    const int*   __restrict__ bits,
    float*       __restrict__ out)
{
    const int p = blockIdx.x * 256 + threadIdx.x;   // pixel id, grid covers kNPix exactly

    const int    b    = p >> 16;                    // p / (H*W)
    const int    hw   = p & 0xFFFF;                 // p % (H*W)
    const size_t base = (size_t)b * (size_t)(kC * kHW) + (size_t)hw;

    const float* __restrict__ fp = features + base;

    // Load all 64 channel values for this pixel into registers.
    // Each iteration is a coalesced 128B/wave global_load_b32 (NT hint).
    float v[kC];
#pragma unroll
    for (int c = 0; c < kC; ++c)
        v[c] = __builtin_nontemporal_load(fp + (size_t)c * kHW);

    // Channel-wise min/max reduction in registers.
    float vmin = v[0], vmax = v[0];
#pragma unroll
    for (int c = 1; c < kC; ++c) {
        vmin = fminf(vmin, v[c]);
        vmax = fmaxf(vmax, v[c]);
    }

    const float rng      = vmax - vmin;
    const bool  mask     = rng > kEps;
    const float rng_safe = mask ? rng : 1.0f;

    const unsigned nb  = (unsigned)bits[p];          // 1..8
    const float    lvl = (float)((1u << nb) - 1u);   // 2^bits - 1, exact in integer

    // Hoist the two divisions; per-element work is then mul/rint/fma/select.
    const float s_fwd = lvl / rng_safe;              // (f - min) * s_fwd  == f_norm * lvl
    const float s_bwd = rng_safe / lvl;              // q * s_bwd + min    == dequant

    float* __restrict__ op = out + base;
#pragma unroll
    for (int c = 0; c < kC; ++c) {
        const float q = rintf((v[c] - vmin) * s_fwd);  // round-half-even == jnp.round
        const float d = fmaf(q, s_bwd, vmin);
        __builtin_nontemporal_store(mask ? d : v[c], op + (size_t)c * kHW);
    }
}

extern "C" void kernel_launch(void* const* d_in, const int* in_sizes, int n_in,
                              void* d_out, int out_size, void* d_ws, size_t ws_size,
                              hipStream_t stream) {
    (void)in_sizes; (void)n_in; (void)d_ws; (void)ws_size; (void)out_size;

    const float* features = (const float*)d_in[0];
    const int*   bits     = (const int*)d_in[1];
    float*       out      = (float*)d_out;

    // kNPix = 1,048,576 pixels, 256 threads (8 wave32) per block -> 4096 blocks.
    dim3 block(256);
    dim3 grid(kNPix / 256);
    AdaptiveQuantizer_463856468234_kernel<<<grid, block, 0, stream>>>(features, bits, out);
}